// GAT_76536317214930
// MI455X (gfx1250) — compile-verified
//
#include <hip/hip_runtime.h>
#include <hip/hip_bf16.h>
#include <math.h>

// ---------------------------------------------------------------------------
// GAT (dead-attention variant):
//   H0 = x @ W0cat            [8192,512] x [512,64]
//   emb0 = elu(adj @ H0)      [8192,8192] x [8192,64]   <- dominant, WMMA f32
//   H1 = emb0 @ W1cat         [8192,64] x [64,64]
//   emb1 = elu(adj @ H1)      dominant, WMMA f32
//   logits = elu(emb1) @ lin_w + lin_b   [8192,64] x [64,40]
//   lsm = log_softmax(logits, axis=1)
// Outputs (concat): lsm[8192*40], emb0[8192*64], emb1[8192*64], logits[8192*40]
// ---------------------------------------------------------------------------

typedef __attribute__((ext_vector_type(2))) float v2f;
typedef __attribute__((ext_vector_type(8))) float v8f;

#define NN     8192
#define NFEAT  512
#define NCLS   40
#define NC     64      // NHEADS * NHID

// ---------------- WMMA tile config for adj @ H -----------------------------
#define BM 32          // rows per block
#define BN 64          // cols per block (= NC)
#define BK 128         // K chunk
#define LDA (BK + 2)   // padded LDS pitch (keeps float2 8B-aligned, avoids conflicts)
#define LDB (BK + 2)

__device__ __forceinline__ float elu1(float x) {
    return x > 0.0f ? x : (__expf(x) - 1.0f);
}

// ---------------------------------------------------------------------------
// emb = elu(adj @ H)   adj: [NN,NN] row-major, H: [NN,64] row-major
// block = 256 threads = 8 waves; wave w -> 16x16 subtile (wR = (w/4)*16, wC = (w%4)*16)
// ---------------------------------------------------------------------------
__global__ __launch_bounds__(256) void adjH_wmma_elu(const float* __restrict__ adj,
                                                     const float* __restrict__ H,
                                                     float* __restrict__ out) {
    __shared__ float sA[BM * LDA];   // adj tile, row-major [BM][BK] (padded)
    __shared__ float sB[BN * LDB];   // H chunk, TRANSPOSED: [col][k] (padded)

    const int tid  = threadIdx.x;
    const int lane = tid & 31;
    const int wave = tid >> 5;
    const int rowBase = blockIdx.x * BM;
    const int wR = (wave >> 2) * 16;   // 0 or 16
    const int wC = (wave & 3) * 16;    // 0,16,32,48

    // ISA fragment indexing (wave32):
    //  A 16x4 f32: lane L holds M = L%16, K = 2*(L/16) + {0,1}  (2 VGPRs, consecutive K)
    //  B 4x16 f32: lane L holds N = L%16, K = 2*(L/16) + {0,1}
    //  C/D 16x16 f32: VGPR v, lane L -> M = v + 8*(L/16), N = L%16
    const int fm   = lane & 15;        // M (A) / N (B) index
    const int koff = (lane >> 4) * 2;  // K sub-offset

    v8f acc = {};

    for (int k0 = 0; k0 < NN; k0 += BK) {
        // ---- stage adj tile [BM x BK], coalesced float4 ----
        for (int i = tid; i < (BM * BK) / 4; i += 256) {
            const int r   = i / (BK / 4);
            const int kk4 = (i % (BK / 4)) * 4;
            const float4 v = *reinterpret_cast<const float4*>(
                &adj[(size_t)(rowBase + r) * NN + k0 + kk4]);
            float* dst = &sA[r * LDA + kk4];
            dst[0] = v.x; dst[1] = v.y; dst[2] = v.z; dst[3] = v.w;
        }
        // ---- stage H chunk [BK x 64] transposed into sB[col][k] ----
        for (int i = tid; i < (BK * BN) / 4; i += 256) {
            const int r  = i / (BN / 4);          // k within chunk
            const int c4 = (i % (BN / 4)) * 4;    // starting col
            const float4 v = *reinterpret_cast<const float4*>(
                &H[(size_t)(k0 + r) * BN + c4]);
            sB[(c4 + 0) * LDB + r] = v.x;
            sB[(c4 + 1) * LDB + r] = v.y;
            sB[(c4 + 2) * LDB + r] = v.z;
            sB[(c4 + 3) * LDB + r] = v.w;
        }
        // prefetch next adj chunk (global_prefetch_b8)
        if (k0 + BK < NN) {
            __builtin_prefetch(&adj[(size_t)(rowBase + (tid >> 3)) * NN + k0 + BK +
                                    ((tid & 7) << 4)], 0, 0);
        }
        __syncthreads();

        const float* aRow = &sA[(wR + fm) * LDA + koff];
        const float* bCol = &sB[(wC + fm) * LDB + koff];
#pragma unroll
        for (int kk = 0; kk < BK; kk += 4) {
            v2f a = *reinterpret_cast<const v2f*>(aRow + kk);
            v2f b = *reinterpret_cast<const v2f*>(bCol + kk);
            acc = __builtin_amdgcn_wmma_f32_16x16x4_f32(
                false, a, false, b, (short)0, acc, false, false);
        }
        __syncthreads();
    }

    // ---- epilogue: ELU + store per C/D layout ----
    const int col  = wC + fm;
    const int rAdd = (lane >> 4) * 8;
#pragma unroll
    for (int v = 0; v < 8; ++v) {
        const int row = rowBase + wR + rAdd + v;
        out[(size_t)row * BN + col] = elu1(acc[v]);
    }
}

// ---------------------------------------------------------------------------
// Small GEMM: out[M,Ncols] = act(A[M,K]) @ W[K,Ncols] (+ bias)
// block = 256 threads: 4 rows x 64 col-slots; A rows staged in LDS.
// ---------------------------------------------------------------------------
__global__ __launch_bounds__(256) void small_gemm(const float* __restrict__ A,
                                                  const float* __restrict__ W,
                                                  const float* __restrict__ bias,
                                                  float* __restrict__ out,
                                                  int K, int Ncols, int applyEluA) {
    __shared__ float sA[4 * 512];
    const int tid = threadIdx.x;
    const int rBase = blockIdx.x * 4;

    for (int i = tid; i < 4 * K; i += 256) {
        const int r = i / K, f = i % K;
        float v = A[(size_t)(rBase + r) * K + f];
        if (applyEluA) v = elu1(v);
        sA[r * K + f] = v;
    }
    __syncthreads();

    const int row = rBase + (tid >> 6);
    const int col = tid & 63;
    if (col < Ncols) {
        const float* ar = &sA[(tid >> 6) * K];
        float acc = 0.0f;
        for (int f = 0; f < K; ++f)
            acc = fmaf(ar[f], W[(size_t)f * Ncols + col], acc);
        if (bias) acc += bias[col];
        out[(size_t)row * Ncols + col] = acc;
    }
}

// ---------------------------------------------------------------------------
// Wcat[f, h*Fout + o] = Wsrc[h, f, o]   (concat heads on output dim)
// ---------------------------------------------------------------------------
__global__ void reshape_heads(const float* __restrict__ Wsrc, float* __restrict__ Wcat,
                              int Hh, int Fin, int Fout) {
    const int idx = blockIdx.x * blockDim.x + threadIdx.x;
    const int tot = Hh * Fin * Fout;
    if (idx >= tot) return;
    const int f = idx / (Hh * Fout);
    const int ho = idx % (Hh * Fout);
    const int h = ho / Fout;
    const int o = ho % Fout;
    Wcat[idx] = Wsrc[((size_t)h * Fin + f) * Fout + o];
}

// ---------------------------------------------------------------------------
// Row-wise log_softmax over 40 classes; one wave32 per row.
// ---------------------------------------------------------------------------
__global__ __launch_bounds__(256) void logsoftmax40(const float* __restrict__ logits,
                                                    float* __restrict__ out, int M) {
    const int wave = blockIdx.x * (blockDim.x >> 5) + (threadIdx.x >> 5);
    const int lane = threadIdx.x & 31;
    if (wave >= M) return;
    const float* r = logits + (size_t)wave * NCLS;
    float v0 = r[lane];                                     // lanes 0..31
    float v1 = (lane < 8) ? r[32 + lane] : -INFINITY;       // lanes 32..39
    float m = fmaxf(v0, v1);
#pragma unroll
    for (int off = 16; off; off >>= 1) m = fmaxf(m, __shfl_xor(m, off, 32));
    float s = __expf(v0 - m) + ((lane < 8) ? __expf(v1 - m) : 0.0f);
#pragma unroll
    for (int off = 16; off; off >>= 1) s += __shfl_xor(s, off, 32);
    const float lse = m + __logf(s);
    float* o = out + (size_t)wave * NCLS;
    o[lane] = v0 - lse;
    if (lane < 8) o[32 + lane] = v1 - lse;
}

// ---------------------------------------------------------------------------
extern "C" void kernel_launch(void* const* d_in, const int* in_sizes, int n_in,
                              void* d_out, int out_size, void* d_ws, size_t ws_size,
                              hipStream_t stream) {
    const float* x     = (const float*)d_in[0];  // [8192,512]
    const float* adj   = (const float*)d_in[1];  // [8192,8192]
    const float* W0    = (const float*)d_in[2];  // [4,512,16]
    const float* W1    = (const float*)d_in[4];  // [4,64,16]
    const float* lin_w = (const float*)d_in[6];  // [64,40]
    const float* lin_b = (const float*)d_in[7];  // [40]

    float* outF = (float*)d_out;
    float* lsm    = outF;                                  // 8192*40
    float* emb0   = outF + (size_t)NN * NCLS;              // 8192*64
    float* emb1   = emb0 + (size_t)NN * NC;                // 8192*64
    float* logits = emb1 + (size_t)NN * NC;                // 8192*40

    float* ws    = (float*)d_ws;
    float* W0cat = ws;                         // 512*64
    float* W1cat = W0cat + (size_t)NFEAT * NC; // 64*64
    float* H0    = W1cat + (size_t)NC * NC;    // 8192*64
    float* H1    = H0 + (size_t)NN * NC;       // 8192*64

    // weight reshapes (head concat)
    reshape_heads<<<(NFEAT * NC + 255) / 256, 256, 0, stream>>>(W0, W0cat, 4, NFEAT, 16);
    reshape_heads<<<(NC * NC + 255) / 256, 256, 0, stream>>>(W1, W1cat, 4, NC, 16);

    // layer 0
    small_gemm<<<NN / 4, 256, 0, stream>>>(x, W0cat, nullptr, H0, NFEAT, NC, 0);
    adjH_wmma_elu<<<NN / BM, 256, 0, stream>>>(adj, H0, emb0);

    // layer 1
    small_gemm<<<NN / 4, 256, 0, stream>>>(emb0, W1cat, nullptr, H1, NC, NC, 0);
    adjH_wmma_elu<<<NN / BM, 256, 0, stream>>>(adj, H1, emb1);

    // classifier + log_softmax
    small_gemm<<<NN / 4, 256, 0, stream>>>(emb1, lin_w, lin_b, logits, NC, NCLS, 1);
    logsoftmax40<<<(NN + 7) / 8, 256, 0, stream>>>(logits, lsm, NN);
}